// YOLOv8_Loss_75557064671787
// MI455X (gfx1250) — compile-verified
//
#include <hip/hip_runtime.h>
#include <math.h>

typedef __attribute__((ext_vector_type(2))) float v2f;
typedef __attribute__((ext_vector_type(8))) float v8f;
typedef __attribute__((ext_vector_type(4))) unsigned int v4u;
typedef __attribute__((ext_vector_type(4))) int v4i;
typedef __attribute__((ext_vector_type(8))) int v8i;

#define A_TOT   8400
#define BSZ     16
#define MB      20
#define NCLS    80
#define CH      144
#define N_BA    (BSZ * A_TOT)      // 134400
#define N_BM    (BSZ * MB)         // 320

__device__ __forceinline__ void levelOf(int a, int& loc, int& W, int& HW, float& st) {
  if (a < 6400)      { loc = a;        W = 80; HW = 6400; st = 8.f;  }
  else if (a < 8000) { loc = a - 6400; W = 40; HW = 1600; st = 16.f; }
  else               { loc = a - 8000; W = 20; HW = 400;  st = 32.f; }
}

__device__ __forceinline__ const float* featOf(int a, const float* f0, const float* f1, const float* f2) {
  return (a < 6400) ? f0 : (a < 8000) ? f1 : f2;
}

// matches reference _ciou exactly (b1-first eps conventions)
__device__ __forceinline__ float ciou_f(const float b1[4], const float b2[4]) {
  const float eps = 1e-7f;
  float w1 = b1[2] - b1[0], h1 = b1[3] - b1[1] + eps;
  float w2 = b2[2] - b2[0], h2 = b2[3] - b2[1] + eps;
  float iw = fmaxf(fminf(b1[2], b2[2]) - fmaxf(b1[0], b2[0]), 0.f);
  float ih = fmaxf(fminf(b1[3], b2[3]) - fmaxf(b1[1], b2[1]), 0.f);
  float inter = iw * ih;
  float uni = w1 * h1 + w2 * h2 - inter + eps;
  float iou = inter / uni;
  float cw = fmaxf(b1[2], b2[2]) - fminf(b1[0], b2[0]);
  float ch = fmaxf(b1[3], b2[3]) - fminf(b1[1], b2[1]);
  float c2 = cw * cw + ch * ch + eps;
  float dx = b2[0] + b2[2] - b1[0] - b1[2];
  float dy = b2[1] + b2[3] - b1[1] - b1[3];
  float rho2 = (dx * dx + dy * dy) * 0.25f;
  float dat = atanf(w2 / h2) - atanf(w1 / h1);
  float v = 0.4052847345693511f * dat * dat;   // 4/pi^2
  float alpha = v / (v - iou + (1.f + eps));
  return iou - (rho2 / c2 + v * alpha);
}

__device__ __forceinline__ float blockReduceSum(float v, float* red) {
  int t = threadIdx.x;
  __syncthreads();
  red[t] = v;
  __syncthreads();
  for (int s = 128; s > 0; s >>= 1) {
    if (t < s) red[t] += red[t + s];
    __syncthreads();
  }
  return red[0];
}

// ---------------- kernels ----------------

__global__ void k_init(int* z, int n) {
  int t = blockIdx.x * 256 + threadIdx.x;
  if (t < n) z[t] = 0;
}

__global__ void k_gt(const float* ann, float* gtbox, int* gtlab, float* gtmask) {
  int t = threadIdx.x;
  if (t >= N_BM) return;
  const float* r = ann + t * 6;
  float cx = r[1], cy = r[2], w = r[3], h = r[4];
  float x1 = (cx - 0.5f * w) * 640.f, y1 = (cy - 0.5f * h) * 640.f;
  float x2 = (cx + 0.5f * w) * 640.f, y2 = (cy + 0.5f * h) * 640.f;
  gtbox[t * 4 + 0] = x1; gtbox[t * 4 + 1] = y1;
  gtbox[t * 4 + 2] = x2; gtbox[t * 4 + 3] = y2;
  gtmask[t] = ((x1 + y1 + x2 + y2) > 0.f) ? 1.f : 0.f;
  gtlab[t]  = (int)r[5];
}

// decode: per-block 32 anchors.
// TDM pulls two 2D tiles (reg 64ch x 32anc, cls 80ch x 32anc) from HBM into
// LDS with pad-per-32-dwords (row stride 33 -> conflict-free banks), tracked
// by TENSORcnt. Softmax(16) per (anchor,side) then dist = probs @ [0..15] via
// 4 chained V_WMMA_F32_16X16X4_F32 per wave (exact fp32). Also accumulates
// softplus over cls logits (BCE base term) from the LDS cls tile.
__global__ __launch_bounds__(256) void k_decode(const float* f0, const float* f1, const float* f2,
                                                float* distw, float* accSP) {
  __shared__ float Pt[64 * 33];   // reg tile   [ch][anchor], stride 33
  __shared__ float Ct[80 * 33];   // cls tile   [cl][anchor], stride 33
  __shared__ float Q[8 * 256];    // 8 waves x (16 rows x 16 bins) probabilities
  __shared__ float red[256];
  int b = blockIdx.y;
  int base = blockIdx.x * 32;     // anchor tile base (grid.x=263, max base 8384)
  int loc0, W, HW; float st;
  levelOf(base, loc0, W, HW, st);
  const float* feat = featOf(base, f0, f1, f2);

#if __has_builtin(__builtin_amdgcn_tensor_load_to_lds) && __has_builtin(__builtin_amdgcn_s_wait_tensorcnt)
  if (threadIdx.x == 0) {
    unsigned ldsP = (unsigned)(unsigned long long)(void*)Pt;
    unsigned ldsC = (unsigned)(unsigned long long)(void*)Ct;
    unsigned long long ga = (unsigned long long)(const void*)(feat + (size_t)b * CH * HW + loc0);
    int tdim0 = HW - loc0;                       // remaining valid anchors (X OOB -> zeros)
    // group1 d0: data_size=4B (2<<16), pad_enable (1<<20), pad_interval=32dw (4<<22), pad_amount=1dw (0<<25)
    int d0 = (2 << 16) | (1 << 20) | (4 << 22);
    v4i z4 = {0, 0, 0, 0};
    // ---- reg tile: tile 32 x 64, tensor stride HW ----
    v4u g0 = { 1u, ldsP, (unsigned)(ga & 0xffffffffull),
               (unsigned)((ga >> 32) & 0x1ffffffull) | (2u << 30) };
    v8i g1 = { d0,
               (int)((tdim0 & 0xffff) << 16),
               (int)(((tdim0 >> 16) & 0xffff) | ((CH & 0xffff) << 16)),
               (int)(32 << 16),          // tile_dim0 = 32 anchors
               64,                        // tile_dim1 = 64 channels
               HW, 0, 0 };                // tensor_dim0_stride = HW
    // ---- cls tile: tile 32 x 80, base advanced by 64 channels ----
    unsigned long long ga2 = ga + (unsigned long long)64 * (unsigned long long)HW * 4ull;
    v4u h0 = { 1u, ldsC, (unsigned)(ga2 & 0xffffffffull),
               (unsigned)((ga2 >> 32) & 0x1ffffffull) | (2u << 30) };
    v8i h1 = { d0,
               (int)((tdim0 & 0xffff) << 16),
               (int)(((tdim0 >> 16) & 0xffff) | ((NCLS & 0xffff) << 16)),
               (int)(32 << 16),
               NCLS,
               HW, 0, 0 };
#if __clang_major__ >= 23
    v8i z8 = {0, 0, 0, 0, 0, 0, 0, 0};
    __builtin_amdgcn_tensor_load_to_lds(g0, g1, z4, z4, z8, 0);
    __builtin_amdgcn_tensor_load_to_lds(h0, h1, z4, z4, z8, 0);
#else
    __builtin_amdgcn_tensor_load_to_lds(g0, g1, z4, z4, 0);
    __builtin_amdgcn_tensor_load_to_lds(h0, h1, z4, z4, 0);
#endif
    __builtin_amdgcn_s_wait_tensorcnt(0);
  }
#else
  for (int i = threadIdx.x; i < CH * 32; i += 256) {
    int chn = i >> 5, la = i & 31;
    float v = (loc0 + la < HW) ? feat[((size_t)b * CH + chn) * HW + loc0 + la] : 0.f;
    if (chn < 64) Pt[chn * 33 + la] = v;
    else          Ct[(chn - 64) * 33 + la] = v;
  }
#endif
  __syncthreads();

  int wave = threadIdx.x >> 5;
  int lane = threadIdx.x & 31;

  // phase 1: softmax of 16 distributions per wave (2 lanes per distribution)
  {
    int row  = lane >> 1;          // 0..15
    int half = lane & 1;
    int bin0 = half * 8;
    int la   = wave * 4 + (row >> 2);
    int side = row & 3;
    float v[8];
#pragma unroll
    for (int i = 0; i < 8; ++i)
      v[i] = Pt[(side * 16 + bin0 + i) * 33 + la];
    float mx = v[0];
#pragma unroll
    for (int i = 1; i < 8; ++i) mx = fmaxf(mx, v[i]);
    mx = fmaxf(mx, __shfl_xor(mx, 1));
    float s = 0.f;
    float e[8];
#pragma unroll
    for (int i = 0; i < 8; ++i) { e[i] = expf(v[i] - mx); s += e[i]; }
    s += __shfl_xor(s, 1);
    float inv = 1.f / s;
#pragma unroll
    for (int i = 0; i < 8; ++i)
      Q[wave * 256 + row * 16 + bin0 + i] = e[i] * inv;
  }
  __syncthreads();

  // phase 2: WMMA expectation. A=16x16 probs, B K-rows = bin weights (all cols equal).
  int m16 = lane & 15;
  int kb2 = (lane >> 4) << 1;      // 0 for lanes 0-15, 2 for lanes 16-31
  v8f dacc = {0.f, 0.f, 0.f, 0.f, 0.f, 0.f, 0.f, 0.f};
#pragma unroll
  for (int c4 = 0; c4 < 4; ++c4) {
    int k = c4 * 4 + kb2;
    v2f a, bb;
    a.x = Q[wave * 256 + m16 * 16 + k];
    a.y = Q[wave * 256 + m16 * 16 + k + 1];
    bb.x = (float)k;
    bb.y = (float)(k + 1);
    dacc = __builtin_amdgcn_wmma_f32_16x16x4_f32(false, a, false, bb, (short)0, dacc, false, false);
  }

  // phase 3: lane 0 holds rows 0-7 (col 0), lane 16 holds rows 8-15 (col 0)
  if (lane == 0 || lane == 16) {
    int rbase = (lane == 0) ? 0 : 8;
#pragma unroll
    for (int v = 0; v < 8; ++v) {
      int r = rbase + v;
      int anc = base + wave * 4 + (r >> 2);
      if (anc < A_TOT)
        distw[((size_t)b * A_TOT + anc) * 4 + (r & 3)] = dacc[v];
    }
  }

  // phase 4: softplus over cls logits of this tile (anchor-major -> conflict-free LDS)
  float sp = 0.f;
  for (int i = threadIdx.x; i < 32 * NCLS; i += 256) {
    int la = i & 31, cl = i >> 5;
    int anc = base + la;
    if (anc < A_TOT) {
      float x = Ct[cl * 33 + la];
      sp += fmaxf(x, 0.f) + log1pf(expf(-fabsf(x)));
    }
  }
  float tot = blockReduceSum(sp, red);
  if (threadIdx.x == 0) atomicAdd(accSP, tot);
}

__global__ void k_box(const float* distw, float* boxw) {
  int t = blockIdx.x * 256 + threadIdx.x;
  if (t >= N_BA) return;
  int a = t % A_TOT;
  int loc, W, HW; float st;
  levelOf(a, loc, W, HW, st);
  float px = (float)(loc % W) + 0.5f;
  float py = (float)(loc / W) + 0.5f;
  const float* d = distw + (size_t)t * 4;
  float* o = boxw + (size_t)t * 4;
  o[0] = px - d[0]; o[1] = py - d[1];
  o[2] = px + d[2]; o[3] = py + d[3];
}

// overlaps & align per (b, gt, anchor)
__global__ void k_align(const float* f0, const float* f1, const float* f2,
                        const float* boxw, const float* gtbox, const int* gtlab,
                        const float* gtmask, float* ovl, float* alg) {
  int bm = blockIdx.x;
  int a  = blockIdx.y * 256 + threadIdx.x;
  if (a >= A_TOT) return;
  int b = bm / MB;
  size_t o = (size_t)bm * A_TOT + a;
  if (gtmask[bm] == 0.f) { ovl[o] = 0.f; alg[o] = 0.f; return; }
  float g[4];
#pragma unroll
  for (int i = 0; i < 4; ++i) g[i] = gtbox[bm * 4 + i];
  int loc, W, HW; float st;
  levelOf(a, loc, W, HW, st);
  float px = ((float)(loc % W) + 0.5f) * st;
  float py = ((float)(loc / W) + 0.5f) * st;
  float mn = fminf(fminf(px - g[0], py - g[1]), fminf(g[2] - px, g[3] - py));
  float mask = (mn > 1e-9f) ? 1.f : 0.f;
  float pb[4];
#pragma unroll
  for (int i = 0; i < 4; ++i) pb[i] = boxw[((size_t)b * A_TOT + a) * 4 + i] * st;
  float cio = ciou_f(g, pb);
  float ov = fmaxf(cio, 0.f) * mask;
  const float* feat = featOf(a, f0, f1, f2);
  float x = feat[((size_t)b * CH + 64 + gtlab[bm]) * HW + loc];
  float sig = 1.f / (1.f + expf(-x));
  float ov2 = ov * ov;
  ovl[o] = ov;
  alg[o] = sqrtf(sig * mask) * (ov2 * ov2 * ov2);
}

// stable top-10 per (b,gt): 10 rounds of (value,index) max-reduction
__global__ __launch_bounds__(256) void k_topk(const float* alg, const float* gtbox,
                                              const float* gtmask, int* fgcnt, unsigned* mpw) {
  __shared__ float rv[256];
  __shared__ int   ri[256];
  __shared__ int   sel[10];
  int bm = blockIdx.x;
  if (gtmask[bm] == 0.f) return;
  int b = bm / MB, m = bm % MB;
  const float* row = alg + (size_t)bm * A_TOT;
  int t = threadIdx.x;
  for (int r = 0; r < 10; ++r) {
    float bv = -1.f; int bi = 1 << 30;
    for (int a = t; a < A_TOT; a += 256) {
      bool used = false;
      for (int j = 0; j < r; ++j) if (sel[j] == a) used = true;
      if (used) continue;
      float v = row[a];
      if (v > bv || (v == bv && a < bi)) { bv = v; bi = a; }
    }
    rv[t] = bv; ri[t] = bi;
    __syncthreads();
    for (int s = 128; s > 0; s >>= 1) {
      if (t < s) {
        float v2 = rv[t + s]; int i2 = ri[t + s];
        if (v2 > rv[t] || (v2 == rv[t] && i2 < ri[t])) { rv[t] = v2; ri[t] = i2; }
      }
      __syncthreads();
    }
    if (t == 0) sel[r] = ri[0];
    __syncthreads();
  }
  if (t < 10) {
    int a = sel[t];
    int loc, W, HW; float st;
    levelOf(a, loc, W, HW, st);
    float px = ((float)(loc % W) + 0.5f) * st;
    float py = ((float)(loc / W) + 0.5f) * st;
    float g0 = gtbox[bm * 4 + 0], g1 = gtbox[bm * 4 + 1];
    float g2 = gtbox[bm * 4 + 2], g3 = gtbox[bm * 4 + 3];
    float mn = fminf(fminf(px - g0, py - g1), fminf(g2 - px, g3 - py));
    if (mn > 1e-9f) {
      atomicAdd(&fgcnt[b * A_TOT + a], 1);
      atomicOr(&mpw[b * A_TOT + a], 1u << m);
    }
  }
}

// resolve multi-assignment; scatter per-gt maxima of align/overlaps
__global__ void k_resolve(const float* ovl, const float* alg, const int* fgcnt,
                          const unsigned* mpw, int* tgw, int* fgw,
                          float* posal, float* posov) {
  int t = blockIdx.x * 256 + threadIdx.x;
  if (t >= N_BA) return;
  int b = t / A_TOT, a = t % A_TOT;
  int c = fgcnt[t];
  unsigned bits = mpw[t];
  int fg = 0, tg = 0;
  if (c > 1) {
    float best = -1.f; int bi = 0;
    for (int m = 0; m < MB; ++m) {
      float v = ovl[(size_t)(b * MB + m) * A_TOT + a];
      if (v > best) { best = v; bi = m; }
    }
    tg = bi; fg = 1;
  } else if (c == 1) {
    tg = __ffs(bits) - 1; fg = 1;
  }
  tgw[t] = tg; fgw[t] = fg;
  if (fg) {
    float av = alg[(size_t)(b * MB + tg) * A_TOT + a];
    float ov = ovl[(size_t)(b * MB + tg) * A_TOT + a];
    atomicMax((int*)&posal[b * MB + tg], __float_as_int(av));  // values >= 0
    atomicMax((int*)&posov[b * MB + tg], __float_as_int(ov));
  }
}

__global__ __launch_bounds__(256) void k_norm(const float* alg, const int* tgw, const int* fgw,
                                              const float* posal, const float* posov,
                                              float* normw, float* accN) {
  __shared__ float red[256];
  int t = blockIdx.x * 256 + threadIdx.x;
  float n = 0.f;
  if (t < N_BA) {
    if (fgw[t]) {
      int b = t / A_TOT, a = t % A_TOT, m = tgw[t];
      float av = alg[(size_t)(b * MB + m) * A_TOT + a];
      n = av * posov[b * MB + m] / (posal[b * MB + m] + 1e-9f);
    }
    normw[t] = n;
  }
  float tot = blockReduceSum(n, red);
  if (threadIdx.x == 0) atomicAdd(accN, tot);
}

// foreground losses: CIoU box loss, DFL, BCE correction (-x*t)
__global__ __launch_bounds__(256) void k_loss(const float* f0, const float* f1, const float* f2,
                                              const float* boxw, const float* gtbox, const int* gtlab,
                                              const int* tgw, const int* fgw, const float* normw,
                                              float* accv) {
  __shared__ float red[256];
  int t = blockIdx.x * 256 + threadIdx.x;
  float lc = 0.f, lb = 0.f, ld = 0.f;
  if (t < N_BA && fgw[t]) {
    int b = t / A_TOT, a = t % A_TOT, m = tgw[t];
    float w = normw[t];
    int loc, W, HW; float st;
    levelOf(a, loc, W, HW, st);
    const float* feat = featOf(a, f0, f1, f2);
    float px = (float)(loc % W) + 0.5f;
    float py = (float)(loc / W) + 0.5f;
    float tb[4];
#pragma unroll
    for (int i = 0; i < 4; ++i) tb[i] = gtbox[(b * MB + m) * 4 + i] / st;
    float pb[4];
#pragma unroll
    for (int i = 0; i < 4; ++i) pb[i] = boxw[(size_t)t * 4 + i];
    lb = (1.f - ciou_f(pb, tb)) * w;
    float xs = feat[((size_t)b * CH + 64 + gtlab[b * MB + m]) * HW + loc];
    lc = -xs * w;
    float tg4[4] = { px - tb[0], py - tb[1], tb[2] - px, tb[3] - py };
    float dsum = 0.f;
#pragma unroll
    for (int side = 0; side < 4; ++side) {
      float tv = fminf(fmaxf(tg4[side], 0.f), 14.99f);
      int   tl = (int)tv;
      float wl = (float)(tl + 1) - tv;
      float lg[16];
      float mx = -1e30f;
#pragma unroll
      for (int k = 0; k < 16; ++k) {
        lg[k] = feat[((size_t)b * CH + side * 16 + k) * HW + loc];
        mx = fmaxf(mx, lg[k]);
      }
      float se = 0.f;
#pragma unroll
      for (int k = 0; k < 16; ++k) se += expf(lg[k] - mx);
      float lse = mx + logf(se);
      float ce_tl = -(lg[tl] - lse);
      float ce_tr = -(lg[tl + 1] - lse);
      dsum += ce_tl * wl + ce_tr * (1.f - wl);
    }
    ld = dsum * 0.25f * w;
  }
  float s = blockReduceSum(lc, red);
  if (threadIdx.x == 0) atomicAdd(&accv[1], s);
  s = blockReduceSum(lb, red);
  if (threadIdx.x == 0) atomicAdd(&accv[2], s);
  s = blockReduceSum(ld, red);
  if (threadIdx.x == 0) atomicAdd(&accv[3], s);
}

__global__ void k_final(const float* accv, float* out) {
  if (threadIdx.x == 0 && blockIdx.x == 0) {
    float tss  = fmaxf(accv[4], 1.f);
    float lbox = accv[2] / tss * 7.5f;
    float lcls = (accv[0] + accv[1]) / tss * 0.5f;
    float ldfl = accv[3] / tss * 1.5f;
    out[0] = lbox + lcls + ldfl;
    out[1] = lbox;
    out[2] = lcls;
    out[3] = ldfl;
  }
}

// ---------------- workspace layout (float words) ----------------
#define OFF_GTBOX  0          // 1280
#define OFF_GTLAB  1280       // 320 (int)
#define OFF_GTMASK 1600       // 320
#define OFF_DIST   1920       // 537600
#define OFF_BOX    539520     // 537600
#define OFF_OVL    1077120    // 2688000
#define OFF_ALG    3765120    // 2688000
#define OFF_FGCNT  6453120    // 134400 (int)
#define OFF_MP     6587520    // 134400 (uint)
#define OFF_TG     6721920    // 134400 (int)
#define OFF_FG     6856320    // 134400 (int)
#define OFF_POSAL  6990720    // 320
#define OFF_POSOV  6991040    // 320
#define OFF_NORM   6991360    // 134400
#define OFF_ACC    7125760    // 8: [0]=softplus [1]=clsCorr [2]=box [3]=dfl [4]=normSum
#define ZERO_N     (7125768 - OFF_FGCNT)   // 672648 words

extern "C" void kernel_launch(void* const* d_in, const int* in_sizes, int n_in,
                              void* d_out, int out_size, void* d_ws, size_t ws_size,
                              hipStream_t stream) {
  const float* f0  = (const float*)d_in[0];
  const float* f1  = (const float*)d_in[1];
  const float* f2  = (const float*)d_in[2];
  const float* ann = (const float*)d_in[3];
  float* ws = (float*)d_ws;

  float*    gtbox  = ws + OFF_GTBOX;
  int*      gtlab  = (int*)(ws + OFF_GTLAB);
  float*    gtmask = ws + OFF_GTMASK;
  float*    distw  = ws + OFF_DIST;
  float*    boxw   = ws + OFF_BOX;
  float*    ovl    = ws + OFF_OVL;
  float*    alg    = ws + OFF_ALG;
  int*      fgcnt  = (int*)(ws + OFF_FGCNT);
  unsigned* mpw    = (unsigned*)(ws + OFF_MP);
  int*      tgw    = (int*)(ws + OFF_TG);
  int*      fgw    = (int*)(ws + OFF_FG);
  float*    posal  = ws + OFF_POSAL;
  float*    posov  = ws + OFF_POSOV;
  float*    normw  = ws + OFF_NORM;
  float*    accv   = ws + OFF_ACC;
  float*    out    = (float*)d_out;

  k_init<<<(ZERO_N + 255) / 256, 256, 0, stream>>>((int*)(ws + OFF_FGCNT), ZERO_N);
  k_gt<<<1, 320, 0, stream>>>(ann, gtbox, gtlab, gtmask);
  k_decode<<<dim3(263, 16), 256, 0, stream>>>(f0, f1, f2, distw, &accv[0]);
  k_box<<<(N_BA + 255) / 256, 256, 0, stream>>>(distw, boxw);
  k_align<<<dim3(320, 33), 256, 0, stream>>>(f0, f1, f2, boxw, gtbox, gtlab, gtmask, ovl, alg);
  k_topk<<<320, 256, 0, stream>>>(alg, gtbox, gtmask, fgcnt, mpw);
  k_resolve<<<(N_BA + 255) / 256, 256, 0, stream>>>(ovl, alg, fgcnt, mpw, tgw, fgw, posal, posov);
  k_norm<<<(N_BA + 255) / 256, 256, 0, stream>>>(alg, tgw, fgw, posal, posov, normw, &accv[4]);
  k_loss<<<(N_BA + 255) / 256, 256, 0, stream>>>(f0, f1, f2, boxw, gtbox, gtlab, tgw, fgw, normw, accv);
  k_final<<<1, 32, 0, stream>>>(accv, out);
}